// GABlock_78675210928560
// MI455X (gfx1250) — compile-verified
//
#include <hip/hip_runtime.h>
#include <math.h>
#include <stdint.h>

// Problem constants (match reference)
#define NB   4
#define LSEQ 512
#define DDIM 128
#define PDIM 64
#define HNUM 8
#define CDIM 32
#define NPTS 8
#define NLROWS (NB*LSEQ)   // 2048

typedef float v2f_t __attribute__((ext_vector_type(2)));
typedef float v8f_t __attribute__((ext_vector_type(8)));

__device__ __forceinline__ v8f_t v8zero() {
  v8f_t v;
#pragma unroll
  for (int i = 0; i < 8; ++i) v[i] = 0.0f;
  return v;
}

// Native fp32 WMMA: D(16x16,f32) = A(16x4,f32) x B(4x16,f32) + C
__device__ __forceinline__ v8f_t wmma_f32(v2f_t a, v2f_t b, v8f_t c) {
  return __builtin_amdgcn_wmma_f32_16x16x4_f32(false, a, false, b, (short)0, c,
                                               false, false);
}

// CDNA5 async global->LDS copy (16B), tracked by ASYNCcnt.
// lds_off: byte offset of destination within LDS (low 32 bits of generic ptr).
__device__ __forceinline__ void async_copy_b128(uint32_t lds_off,
                                                const float* gaddr) {
  asm volatile("global_load_async_to_lds_b128 %0, %1, off"
               :
               : "v"(lds_off), "v"(gaddr)
               : "memory");
}

// ---- wave32 reductions -----------------------------------------------------
__device__ __forceinline__ float warpMax(float v) {
#pragma unroll
  for (int o = 16; o > 0; o >>= 1) v = fmaxf(v, __shfl_xor(v, o, 32));
  return v;
}
__device__ __forceinline__ float warpSum(float v) {
#pragma unroll
  for (int o = 16; o > 0; o >>= 1) v += __shfl_xor(v, o, 32);
  return v;
}
// 256-thread (8-wave) block reductions; s8 is shared float[8]
__device__ __forceinline__ float blockMax(float v, float* s8) {
  v = warpMax(v);
  __syncthreads();
  if ((threadIdx.x & 31) == 0) s8[threadIdx.x >> 5] = v;
  __syncthreads();
  float r = s8[0];
#pragma unroll
  for (int i = 1; i < 8; ++i) r = fmaxf(r, s8[i]);
  return r;
}
__device__ __forceinline__ float blockSum(float v, float* s8) {
  v = warpSum(v);
  __syncthreads();
  if ((threadIdx.x & 31) == 0) s8[threadIdx.x >> 5] = v;
  __syncthreads();
  float r = 0.f;
#pragma unroll
  for (int i = 0; i < 8; ++i) r += s8[i];
  return r;
}

// ---------------------------------------------------------------------------
// Generic fp32 WMMA GEMM: C(MxN) = A(MxK) * B(KxN), row-major, all dims %16/%4.
// One wave per 16x16 C tile; 8 waves per block.
// ---------------------------------------------------------------------------
__global__ __launch_bounds__(256) void k_gemm(const float* __restrict__ A,
                                              const float* __restrict__ B,
                                              float* __restrict__ C, int M,
                                              int Nn, int K) {
  const int lane = threadIdx.x & 31, w = threadIdx.x >> 5;
  const int tile = blockIdx.x * 8 + w;
  const int ntn = Nn >> 4;
  const int tm = tile / ntn, tn = tile % ntn;
  if (tm * 16 >= M) return;
  const int rc = lane & 15;          // A row / B col for this lane
  const int kk = (lane >> 4) * 2;    // K sub-pair held by this lane
  v8f_t acc = v8zero();
  for (int k0 = 0; k0 < K; k0 += 4) {
    v2f_t a, b;
    a.x = A[(size_t)(tm * 16 + rc) * K + k0 + kk];
    a.y = A[(size_t)(tm * 16 + rc) * K + k0 + kk + 1];
    b.x = B[(size_t)(k0 + kk) * Nn + tn * 16 + rc];
    b.y = B[(size_t)(k0 + kk + 1) * Nn + tn * 16 + rc];
    acc = wmma_f32(a, b, acc);
  }
#pragma unroll
  for (int ii = 0; ii < 8; ++ii) {
    int r = ii + ((lane >> 4) << 3);
    C[(size_t)(tm * 16 + r) * Nn + tn * 16 + (lane & 15)] = acc[ii];
  }
}

// ---------------------------------------------------------------------------
// In-place local->global frame transform of H*NP points per residue.
// p layout: (N*L, 64, 3) ; out_i = R[i][j]*p[j] + t[i]
// ---------------------------------------------------------------------------
__global__ void k_frames(float* __restrict__ p, const float* __restrict__ R,
                         const float* __restrict__ tv, int total) {
  int idx = blockIdx.x * blockDim.x + threadIdx.x;
  if (idx >= total) return;
  int nl = idx >> 6;
  float* pp = p + (size_t)idx * 3;
  const float* Rm = R + (size_t)nl * 9;
  const float* t3 = tv + (size_t)nl * 3;
  float a = pp[0], b = pp[1], c = pp[2];
  pp[0] = Rm[0] * a + Rm[1] * b + Rm[2] * c + t3[0];
  pp[1] = Rm[3] * a + Rm[4] * b + Rm[5] * c + t3[1];
  pp[2] = Rm[6] * a + Rm[7] * b + Rm[8] * c + t3[2];
}

// Sum of squares over the 24 components of one head's points: out (N*L, 8)
__global__ void k_sumsq(const float* __restrict__ p, float* __restrict__ out,
                        int total) {
  int idx = blockIdx.x * blockDim.x + threadIdx.x;
  if (idx >= total) return;
  const float* base = p + (size_t)(idx >> 3) * 192 + (idx & 7) * 24;
  float s = 0.f;
#pragma unroll
  for (int i = 0; i < 24; ++i) s += base[i] * base[i];
  out[idx] = s;
}

__global__ void k_gamma(const float* __restrict__ sc, float* __restrict__ g) {
  int h = threadIdx.x;
  if (h < HNUM) {
    float c = sc[h];
    g[h] = (c > 0.f) ? (c + log1pf(expf(-c))) : log1pf(expf(c));
  }
}

__global__ __launch_bounds__(256) void k_ratio(
    const unsigned char* __restrict__ intra,
    const unsigned char* __restrict__ inter, float* __restrict__ ratio) {
  __shared__ float s1[8], s2[8];
  const int n = blockIdx.x, t = threadIdx.x;
  const unsigned char* ip = intra + (size_t)n * LSEQ * LSEQ;
  const unsigned char* ep = inter + (size_t)n * LSEQ * LSEQ;
  float si = 0.f, se = 0.f;
  for (int i = t; i < LSEQ * LSEQ; i += 256) {
    si += ip[i] ? 1.f : 0.f;
    se += ep[i] ? 1.f : 0.f;
  }
  si = warpSum(si);
  se = warpSum(se);
  if ((t & 31) == 0) { s1[t >> 5] = si; s2[t >> 5] = se; }
  __syncthreads();
  if (t == 0) {
    float a = 0.f, b = 0.f;
#pragma unroll
    for (int i = 0; i < 8; ++i) { a += s1[i]; b += s2[i]; }
    ratio[n] = a / (a + b);
  }
}

// ---------------------------------------------------------------------------
// Logits: per (n, 16-query tile). 8 waves == 8 heads (WMMA QK^T + point dot),
// all 256 threads cooperate on z*Wpair. z is streamed through a double-
// buffered LDS pipeline with CDNA5 async global->LDS copies (ASYNCcnt).
// logits layout (N,L,L,H), scaled by sqrt(1/3).
// ---------------------------------------------------------------------------
__global__ __launch_bounds__(256) void k_logits(
    const float* __restrict__ qn, const float* __restrict__ kn,
    const float* __restrict__ qp, const float* __restrict__ kp,
    const float* __restrict__ qp2, const float* __restrict__ kp2,
    const float* __restrict__ z, const float* __restrict__ Wpair,
    const float* __restrict__ gam, float* __restrict__ logits) {
  const int n = blockIdx.y, l0 = blockIdx.x * 16;
  const int t = threadIdx.x, lane = t & 31, h = t >> 5;
  const int pi = t >> 4, pj = t & 15;  // (query,key) pair for z-dot work
  __shared__ float zbuf[2][16 * 16 * 16];  // 2 x 16 KB ping-pong z chunks
  __shared__ float pairbuf[16 * 16 * 8];   // 8 KB pair logits
  const int rc = lane & 15;
  const int kk = (lane >> 4) * 2;
  const float rsC = 0.17677669529663687f;   // 1/sqrt(32)
  const float fin = 0.5773502691896258f;    // sqrt(1/3)
  for (int mt = 0; mt < 32; ++mt) {
    const int m0 = mt * 16;
    float pr[8];
#pragma unroll
    for (int hh = 0; hh < 8; ++hh) pr[hh] = 0.f;
    for (int pc = 0; pc < 4; ++pc) {
      __syncthreads();  // target ping-pong buffer free (prior reads done)
      // issue async chunk(s): chunk pc on first iter, then chunk pc+1
      if (pc == 0) {
#pragma unroll
        for (int rr = 0; rr < 4; ++rr) {
          int f16 = t + rr * 256;  // 16-byte unit index within chunk
          int pq = (f16 & 3) * 4, jj = (f16 >> 2) & 15, ii = f16 >> 6;
          const float* g =
              &z[(((size_t)(n * LSEQ + l0 + ii) * LSEQ) + m0 + jj) * PDIM + pq];
          async_copy_b128((uint32_t)(uintptr_t)&zbuf[0][f16 * 4], g);
        }
      }
      if (pc < 3) {
        const int nc = pc + 1;
#pragma unroll
        for (int rr = 0; rr < 4; ++rr) {
          int f16 = t + rr * 256;
          int pq = (f16 & 3) * 4, jj = (f16 >> 2) & 15, ii = f16 >> 6;
          const float* g =
              &z[(((size_t)(n * LSEQ + l0 + ii) * LSEQ) + m0 + jj) * PDIM +
                 nc * 16 + pq];
          async_copy_b128((uint32_t)(uintptr_t)&zbuf[nc & 1][f16 * 4], g);
        }
        asm volatile("s_wait_asynccnt 0x4" ::: "memory");  // chunk pc landed
      } else {
        asm volatile("s_wait_asynccnt 0x0" ::: "memory");  // drain last chunk
      }
      __syncthreads();  // all waves observe chunk pc in LDS
      const float* zrow = &zbuf[pc & 1][(pi * 16 + pj) * 16];
#pragma unroll
      for (int p = 0; p < 16; ++p) {
        float zv = zrow[p];
        const float* wp = &Wpair[(pc * 16 + p) * HNUM];
#pragma unroll
        for (int hh = 0; hh < 8; ++hh) pr[hh] += zv * wp[hh];
      }
    }
#pragma unroll
    for (int hh = 0; hh < 8; ++hh) pairbuf[(pi * 16 + pj) * 8 + hh] = pr[hh];

    // WMMA: node logits (K=32) and point dot (K=24) for this head
    v8f_t an = v8zero(), asp = v8zero();
    const float* qb = qn + ((size_t)(n * LSEQ) + l0) * 256 + h * CDIM;
    const float* kb = kn + ((size_t)(n * LSEQ) + m0) * 256 + h * CDIM;
    for (int k0 = 0; k0 < CDIM; k0 += 4) {
      v2f_t a, b;
      a.x = qb[rc * 256 + k0 + kk];  a.y = qb[rc * 256 + k0 + kk + 1];
      b.x = kb[rc * 256 + k0 + kk];  b.y = kb[rc * 256 + k0 + kk + 1];
      an = wmma_f32(a, b, an);
    }
    const float* qpb = qp + ((size_t)(n * LSEQ) + l0) * 192 + h * 24;
    const float* kpb = kp + ((size_t)(n * LSEQ) + m0) * 192 + h * 24;
    for (int k0 = 0; k0 < 24; k0 += 4) {
      v2f_t a, b;
      a.x = qpb[rc * 192 + k0 + kk];  a.y = qpb[rc * 192 + k0 + kk + 1];
      b.x = kpb[rc * 192 + k0 + kk];  b.y = kpb[rc * 192 + k0 + kk + 1];
      asp = wmma_f32(a, b, asp);
    }
    __syncthreads();
    const float g = gam[h];
#pragma unroll
    for (int ii = 0; ii < 8; ++ii) {
      int r = ii + ((lane >> 4) << 3);
      int l = l0 + r, m = m0 + (lane & 15);
      float node = an[ii] * rsC;
      float pairv = pairbuf[(r * 16 + (lane & 15)) * 8 + h];
      float ssq = qp2[(n * LSEQ + l) * 8 + h] + kp2[(n * LSEQ + m) * 8 + h] -
                  2.f * asp[ii];
      float sp = -g * ssq * (1.f / 12.f);  // gamma * sqrt(2/(9*NP))/2 = g/12
      logits[(((size_t)(n * LSEQ + l) * LSEQ) + m) * 8 + h] =
          (node + pairv + sp) * fin;
    }
  }
}

// ---------------------------------------------------------------------------
// Dual-mask softmax blend, in place on logits -> alpha. One block per (n,l).
// ---------------------------------------------------------------------------
__global__ __launch_bounds__(256) void k_softmax(
    float* __restrict__ logits, const unsigned char* __restrict__ intra,
    const unsigned char* __restrict__ inter, const float* __restrict__ ratio) {
  __shared__ float s8[8];
  const int n = blockIdx.y, l = blockIdx.x, t = threadIdx.x;
  float* rowp = logits + ((size_t)(n * LSEQ + l) * LSEQ) * 8;
  const unsigned char* im = intra + ((size_t)(n * LSEQ + l)) * LSEQ;
  const unsigned char* em = inter + ((size_t)(n * LSEQ + l)) * LSEQ;
  const int m0 = t, m1 = t + 256;
  float lg[2][8], acc[2][8];
#pragma unroll
  for (int hh = 0; hh < 8; ++hh) {
    lg[0][hh] = rowp[m0 * 8 + hh];
    lg[1][hh] = rowp[m1 * 8 + hh];
    acc[0][hh] = 0.f;
    acc[1][hh] = 0.f;
  }
  const float r = ratio[n];
  for (int mid = 0; mid < 2; ++mid) {
    const unsigned char k0 = (mid == 0) ? im[m0] : em[m0];
    const unsigned char k1 = (mid == 0) ? im[m1] : em[m1];
    const float coef = (mid == 0) ? r : (1.f - r);
#pragma unroll
    for (int hh = 0; hh < 8; ++hh) {
      float sh0 = k0 ? lg[0][hh] : lg[0][hh] - 1e5f;
      float sh1 = k1 ? lg[1][hh] : lg[1][hh] - 1e5f;
      float mx = blockMax(fmaxf(sh0, sh1), s8);
      float e0 = expf(sh0 - mx), e1 = expf(sh1 - mx);
      float S = blockSum(e0 + e1, s8);
      float inv = coef / S;
      if (k0) acc[0][hh] += e0 * inv;
      if (k1) acc[1][hh] += e1 * inv;
    }
  }
#pragma unroll
  for (int hh = 0; hh < 8; ++hh) {
    rowp[m0 * 8 + hh] = acc[0][hh];
    rowp[m1 * 8 + hh] = acc[1][hh];
  }
}

// ---------------------------------------------------------------------------
// Aggregation: per (n, 16-query tile). Wave h: WMMA alpha@vn (K=512) and
// alpha@vp_global; all threads: VALU alpha-weighted z; frame back-transform.
// agg layout per row: [node 0..255 | pair 256..767 | spatial 768..831]
// ---------------------------------------------------------------------------
__global__ __launch_bounds__(256) void k_agg(
    const float* __restrict__ alpha, const float* __restrict__ vn,
    const float* __restrict__ vp, const float* __restrict__ z,
    const float* __restrict__ R, const float* __restrict__ tv,
    float* __restrict__ agg) {
  const int n = blockIdx.y, l0 = blockIdx.x * 16;
  const int t = threadIdx.x, lane = t & 31, h = t >> 5;
  __shared__ float fsbuf[16 * 8 * 24];  // 12 KB: global-frame point sums
  const int rc = lane & 15;
  const int kk = (lane >> 4) * 2;
  // ---- feat_node: alpha(16x512) @ vn_head(512x32) ----
  for (int ct = 0; ct < 2; ++ct) {
    v8f_t acc = v8zero();
    for (int m0 = 0; m0 < LSEQ; m0 += 4) {
      v2f_t a, b;
      a.x = alpha[((size_t)(n * LSEQ + l0 + rc) * LSEQ + m0 + kk) * 8 + h];
      a.y = alpha[((size_t)(n * LSEQ + l0 + rc) * LSEQ + m0 + kk + 1) * 8 + h];
      b.x = vn[(size_t)(n * LSEQ + m0 + kk) * 256 + h * CDIM + ct * 16 + rc];
      b.y = vn[(size_t)(n * LSEQ + m0 + kk + 1) * 256 + h * CDIM + ct * 16 + rc];
      acc = wmma_f32(a, b, acc);
    }
#pragma unroll
    for (int ii = 0; ii < 8; ++ii) {
      int r = ii + ((lane >> 4) << 3);
      agg[(size_t)(n * LSEQ + l0 + r) * 832 + h * CDIM + ct * 16 + (lane & 15)] =
          acc[ii];
    }
  }
  // ---- feat_spatial (global frame): alpha(16x512) @ vp_head(512x24) ----
  for (int ct = 0; ct < 2; ++ct) {
    int cidx = ct * 16 + rc;
    int csafe = (cidx < 24) ? cidx : (16 + (cidx & 7));  // avoid OOB, unused
    v8f_t acc = v8zero();
    for (int m0 = 0; m0 < LSEQ; m0 += 4) {
      v2f_t a, b;
      a.x = alpha[((size_t)(n * LSEQ + l0 + rc) * LSEQ + m0 + kk) * 8 + h];
      a.y = alpha[((size_t)(n * LSEQ + l0 + rc) * LSEQ + m0 + kk + 1) * 8 + h];
      b.x = vp[(size_t)(n * LSEQ + m0 + kk) * 192 + h * 24 + csafe];
      b.y = vp[(size_t)(n * LSEQ + m0 + kk + 1) * 192 + h * 24 + csafe];
      acc = wmma_f32(a, b, acc);
    }
#pragma unroll
    for (int ii = 0; ii < 8; ++ii) {
      int r = ii + ((lane >> 4) << 3);
      int c = ct * 16 + (lane & 15);
      if (c < 24) fsbuf[(r * 8 + h) * 24 + c] = acc[ii];
    }
  }
  __syncthreads();
  // ---- feat_pair: out[l,h,p] = sum_m alpha[l,m,h] * z[l,m,p] (VALU) ----
  {
    const int p = t & 63, lsub = t >> 6;
    for (int lgp = 0; lgp < 4; ++lgp) {
      const int li = lgp * 4 + lsub, l = l0 + li;
      const float* arow = &alpha[((size_t)(n * LSEQ + l) * LSEQ) * 8];
      const float* zrow = &z[((size_t)(n * LSEQ + l) * LSEQ) * PDIM];
      float a8[8];
#pragma unroll
      for (int hh = 0; hh < 8; ++hh) a8[hh] = 0.f;
      for (int m = 0; m < LSEQ; ++m) {
        float zv = zrow[(size_t)m * PDIM + p];
        const float4* av = (const float4*)&arow[m * 8];
        float4 a0 = av[0], a1 = av[1];
        a8[0] += a0.x * zv; a8[1] += a0.y * zv;
        a8[2] += a0.z * zv; a8[3] += a0.w * zv;
        a8[4] += a1.x * zv; a8[5] += a1.y * zv;
        a8[6] += a1.z * zv; a8[7] += a1.w * zv;
      }
#pragma unroll
      for (int hh = 0; hh < 8; ++hh)
        agg[(size_t)(n * LSEQ + l) * 832 + 256 + hh * PDIM + p] = a8[hh];
    }
  }
  // ---- spatial: back to local frame + norm ----
  for (int idx = t; idx < 16 * 8 * 8; idx += 256) {
    int li = idx >> 6, hh = (idx >> 3) & 7, p = idx & 7;
    int l = l0 + li;
    const float* g3 = &fsbuf[(li * 8 + hh) * 24 + p * 3];
    const float* Rm = &R[(size_t)(n * LSEQ + l) * 9];
    const float* t3 = &tv[(size_t)(n * LSEQ + l) * 3];
    float v0 = g3[0] - t3[0], v1 = g3[1] - t3[1], v2 = g3[2] - t3[2];
    float x0 = Rm[0] * v0 + Rm[3] * v1 + Rm[6] * v2;
    float x1 = Rm[1] * v0 + Rm[4] * v1 + Rm[7] * v2;
    float x2 = Rm[2] * v0 + Rm[5] * v1 + Rm[8] * v2;
    agg[(size_t)(n * LSEQ + l) * 832 + 768 + hh * NPTS + p] =
        sqrtf(x0 * x0 + x1 * x1 + x2 * x2);
  }
}

// ---------------------------------------------------------------------------
// Fused transitions: feats = LN(x + agg@W1 + b1); 3-layer MLP; LN(feats+out).
// One block per 16-row tile; 8 waves cover 128 output cols; A from LDS.
// ---------------------------------------------------------------------------
__device__ __forceinline__ void gemm128_lds(const float* __restrict__ Asm,
                                            const float* __restrict__ B,
                                            const float* __restrict__ bias,
                                            const float* __restrict__ resid,
                                            float* __restrict__ Osm, bool relu,
                                            int lane, int w) {
  const int rc = lane & 15, kk = (lane >> 4) * 2;
  v8f_t acc = v8zero();
  for (int k0 = 0; k0 < DDIM; k0 += 4) {
    v2f_t a, b;
    a.x = Asm[rc * DDIM + k0 + kk];
    a.y = Asm[rc * DDIM + k0 + kk + 1];
    b.x = B[(k0 + kk) * DDIM + w * 16 + rc];
    b.y = B[(k0 + kk + 1) * DDIM + w * 16 + rc];
    acc = wmma_f32(a, b, acc);
  }
#pragma unroll
  for (int ii = 0; ii < 8; ++ii) {
    int r = ii + ((lane >> 4) << 3);
    int c = w * 16 + (lane & 15);
    float v = acc[ii] + bias[c];
    if (resid) v += resid[r * DDIM + c];
    Osm[r * DDIM + c] = relu ? fmaxf(v, 0.f) : v;
  }
}

__device__ __forceinline__ void ln_row(float* rp, const float* g,
                                       const float* b, float* outp) {
  float m = 0.f;
  for (int c = 0; c < DDIM; ++c) m += rp[c];
  m *= (1.f / DDIM);
  float v = 0.f;
  for (int c = 0; c < DDIM; ++c) { float d = rp[c] - m; v += d * d; }
  v *= (1.f / DDIM);
  float rs = rsqrtf(v + 1e-5f);
  for (int c = 0; c < DDIM; ++c) outp[c] = (rp[c] - m) * rs * g[c] + b[c];
}

__global__ __launch_bounds__(256) void k_trans(
    const float* __restrict__ agg, const float* __restrict__ x,
    const float* __restrict__ W1, const float* __restrict__ b1,
    const float* __restrict__ W2a, const float* __restrict__ b2a,
    const float* __restrict__ W2b, const float* __restrict__ b2b,
    const float* __restrict__ W2c, const float* __restrict__ b2c,
    const float* __restrict__ ln1g, const float* __restrict__ ln1b,
    const float* __restrict__ ln2g, const float* __restrict__ ln2b,
    float* __restrict__ out) {
  const int r0 = blockIdx.x * 16;
  const int t = threadIdx.x, lane = t & 31, w = t >> 5;
  __shared__ float sF[16 * DDIM];  // feats
  __shared__ float sA[16 * DDIM];  // hidden 1 / tmp
  __shared__ float sB[16 * DDIM];  // hidden 2
  const int rc = lane & 15, kk = (lane >> 4) * 2;
  // feats_raw = x + agg @ W1 + b1   (K = 832)
  {
    v8f_t acc = v8zero();
    for (int k0 = 0; k0 < 832; k0 += 4) {
      v2f_t a, b;
      a.x = agg[(size_t)(r0 + rc) * 832 + k0 + kk];
      a.y = agg[(size_t)(r0 + rc) * 832 + k0 + kk + 1];
      b.x = W1[(size_t)(k0 + kk) * DDIM + w * 16 + rc];
      b.y = W1[(size_t)(k0 + kk + 1) * DDIM + w * 16 + rc];
      acc = wmma_f32(a, b, acc);
    }
#pragma unroll
    for (int ii = 0; ii < 8; ++ii) {
      int r = ii + ((lane >> 4) << 3);
      int c = w * 16 + (lane & 15);
      sF[r * DDIM + c] = acc[ii] + b1[c] + x[(size_t)(r0 + r) * DDIM + c];
    }
  }
  __syncthreads();
  if (t < 16) ln_row(&sF[t * DDIM], ln1g, ln1b, &sF[t * DDIM]);
  __syncthreads();
  gemm128_lds(sF, W2a, b2a, nullptr, sA, true, lane, w);
  __syncthreads();
  gemm128_lds(sA, W2b, b2b, nullptr, sB, true, lane, w);
  __syncthreads();
  gemm128_lds(sB, W2c, b2c, sF, sA, false, lane, w);  // feats + out
  __syncthreads();
  if (t < 16) ln_row(&sA[t * DDIM], ln2g, ln2b, &out[(size_t)(r0 + t) * DDIM]);
}

// ---------------------------------------------------------------------------
extern "C" void kernel_launch(void* const* d_in, const int* in_sizes, int n_in,
                              void* d_out, int out_size, void* d_ws,
                              size_t ws_size, hipStream_t stream) {
  (void)in_sizes; (void)n_in; (void)out_size; (void)ws_size;
  const float* R  = (const float*)d_in[0];
  const float* tv = (const float*)d_in[1];
  const float* x  = (const float*)d_in[2];
  const float* z  = (const float*)d_in[3];
  const unsigned char* intra = (const unsigned char*)d_in[4];
  const unsigned char* inter = (const unsigned char*)d_in[5];
  const float* Wq = (const float*)d_in[6];
  const float* Wk = (const float*)d_in[7];
  const float* Wv = (const float*)d_in[8];
  const float* Wpair = (const float*)d_in[9];
  const float* sc = (const float*)d_in[10];
  const float* Wqp = (const float*)d_in[11];
  const float* Wkp = (const float*)d_in[12];
  const float* Wvp = (const float*)d_in[13];
  const float* W1 = (const float*)d_in[14];
  const float* b1 = (const float*)d_in[15];
  const float* W2a = (const float*)d_in[16];
  const float* b2a = (const float*)d_in[17];
  const float* W2b = (const float*)d_in[18];
  const float* b2b = (const float*)d_in[19];
  const float* W2c = (const float*)d_in[20];
  const float* b2c = (const float*)d_in[21];
  const float* ln1g = (const float*)d_in[22];
  const float* ln1b = (const float*)d_in[23];
  const float* ln2g = (const float*)d_in[24];
  const float* ln2b = (const float*)d_in[25];

  // ---- workspace layout (floats) ----
  float* ws = (float*)d_ws;
  size_t off = 0;
  float* qn = ws + off; off += (size_t)NLROWS * 256;
  float* kn = ws + off; off += (size_t)NLROWS * 256;
  float* vn = ws + off; off += (size_t)NLROWS * 256;
  float* qp = ws + off; off += (size_t)NLROWS * 192 + 16;
  float* kp = ws + off; off += (size_t)NLROWS * 192 + 16;
  float* vp = ws + off; off += (size_t)NLROWS * 192 + 16;
  float* qp2 = ws + off; off += (size_t)NLROWS * 8;
  float* kp2 = ws + off; off += (size_t)NLROWS * 8;
  float* gam = ws + off; off += 8;
  float* ratio = ws + off; off += 8;
  float* logits = ws + off; off += (size_t)NLROWS * LSEQ * 8;  // -> alpha
  float* agg = ws + off; off += (size_t)NLROWS * 832;

  // ---- projections (fp32 WMMA GEMMs) ----
  k_gemm<<<256, 256, 0, stream>>>(x, Wq, qn, NLROWS, 256, DDIM);
  k_gemm<<<256, 256, 0, stream>>>(x, Wk, kn, NLROWS, 256, DDIM);
  k_gemm<<<256, 256, 0, stream>>>(x, Wv, vn, NLROWS, 256, DDIM);
  k_gemm<<<192, 256, 0, stream>>>(x, Wqp, qp, NLROWS, 192, DDIM);
  k_gemm<<<192, 256, 0, stream>>>(x, Wkp, kp, NLROWS, 192, DDIM);
  k_gemm<<<192, 256, 0, stream>>>(x, Wvp, vp, NLROWS, 192, DDIM);
  // ---- frames + precomputes ----
  k_frames<<<512, 256, 0, stream>>>(qp, R, tv, NLROWS * 64);
  k_frames<<<512, 256, 0, stream>>>(kp, R, tv, NLROWS * 64);
  k_frames<<<512, 256, 0, stream>>>(vp, R, tv, NLROWS * 64);
  k_sumsq<<<64, 256, 0, stream>>>(qp, qp2, NLROWS * 8);
  k_sumsq<<<64, 256, 0, stream>>>(kp, kp2, NLROWS * 8);
  k_gamma<<<1, 32, 0, stream>>>(sc, gam);
  k_ratio<<<NB, 256, 0, stream>>>(intra, inter, ratio);
  // ---- attention ----
  k_logits<<<dim3(LSEQ / 16, NB), 256, 0, stream>>>(qn, kn, qp, kp, qp2, kp2,
                                                    z, Wpair, gam, logits);
  k_softmax<<<dim3(LSEQ, NB), 256, 0, stream>>>(logits, intra, inter, ratio);
  k_agg<<<dim3(LSEQ / 16, NB), 256, 0, stream>>>(logits, vn, vp, z, R, tv, agg);
  // ---- transitions ----
  k_trans<<<NLROWS / 16, 256, 0, stream>>>(agg, x, W1, b1, W2a, b2a, W2b, b2b,
                                           W2c, b2c, ln1g, ln1b, ln2g, ln2b,
                                           (float*)d_out);
}